// SelfAttentionModule_26113401160242
// MI455X (gfx1250) — compile-verified
//
#include <hip/hip_runtime.h>
#include <hip/hip_bf16.h>

// ---------------------------------------------------------------------------
// Local 7x7 self-attention (Q=K=V=x), x:(4,256,64,64) fp32.
// Per 8x8 query tile:
//   - stage 14x14x256 halo tile into LDS channel-major via the Tensor Data
//     Mover (interior tiles: 1 tensor_load_to_lds; border tiles: scalar path)
//   - S = K^T Q  (fp32 WMMA 16x16x4)
//   - masked softmax over the 49 in-window taps
//   - Out = V P  (fp32 WMMA 16x16x4)
// Memory-bound: ~33.5 MB HBM traffic -> ~1.4 us floor @ 23.3 TB/s; WMMA work
// (~3.3 GFLOP of GEMM) is negligible against any matrix-pipe ceiling, so fp32
// WMMA keeps reference precision at zero cost.
// ---------------------------------------------------------------------------

typedef float v2f  __attribute__((ext_vector_type(2)));
typedef float v8f  __attribute__((ext_vector_type(8)));
typedef unsigned int u32x4 __attribute__((ext_vector_type(4)));
typedef int  i32x4 __attribute__((ext_vector_type(4)));
typedef int  i32x8 __attribute__((ext_vector_type(8)));

#define BATCH    4
#define CH       256
#define HH       64
#define WW       64
#define KS       7
#define PADW     3
#define QT       8            // query tile side
#define ET       14           // extended (halo) tile side = QT + KS - 1
#define MKEYS    196          // ET*ET key positions (also LDS plane stride; 196%64==4 -> bank-clean)
#define MPAD     208          // GEMM M/K extent rounded to multiple of 16
#define NQ       64           // QT*QT queries per tile
#define SSTRIDE  68           // LDS row stride (floats) for S/P   [MPAD][SSTRIDE]
#define NTHREADS 256
#define NWAVES   8            // wave32

// keys: channel-major [CH][MKEYS]; sp: [MPAD][SSTRIDE]; + per-query stats
#define SMEM_FLOATS (CH * MKEYS + MPAD * SSTRIDE + 2 * NQ)

__global__ __launch_bounds__(NTHREADS)
void local_attn_wmma_f32(const float* __restrict__ x, float* __restrict__ out)
{
    extern __shared__ float smem[];
    float* keys = smem;                          // [CH][MKEYS]   halo tile, channel-major
    float* sp   = smem + CH * MKEYS;             // [MPAD][SSTRIDE]  S then P
    float* mxv  = sp + MPAD * SSTRIDE;           // [NQ] per-query max
    float* rsum = mxv + NQ;                      // [NQ] per-query 1/sum

    const int t    = threadIdx.x;
    const int lane = t & 31;
    const int wid  = t >> 5;
    const int tx   = blockIdx.x;                 // tile x (0..7)
    const int ty   = blockIdx.y;                 // tile y (0..7)
    const int b    = blockIdx.z;                 // batch

    const int gx0 = tx * QT - PADW;
    const int gy0 = ty * QT - PADW;
    const float* xb = x + (size_t)b * CH * HH * WW;

    const bool interior = (tx >= 1) & (tx <= 6) & (ty >= 1) & (ty <= 6);

    if (interior) {
        // ---- TDM: one DMA for the whole 14x14x256 halo tile, global -> LDS.
        //      3-D descriptor: x (14, contiguous), y (14, stride 64), c (256,
        //      stride 4096). tensor_dim = remaining extent from tile start
        //      (right/bottom OOB would zero-fill; interior => never hit).
        if (wid == 0) {
            const unsigned lds_off = (unsigned)(uintptr_t)(void*)keys;  // low 32b of generic ptr = LDS offset
            const unsigned long long ga =
                (unsigned long long)(uintptr_t)(xb + gy0 * WW + gx0);
            const unsigned ga_lo = (unsigned)ga;
            const unsigned ga_hi = (unsigned)(ga >> 32) & 0x01FFFFFFu;
            const unsigned tdim0 = (unsigned)(WW - gx0);   // remaining width
            const unsigned tdim1 = (unsigned)(HH - gy0);   // remaining height

            u32x4 g0 = { 1u,                     // count=1, is_restore=0, gather off
                         lds_off,                // lds_addr (bytes)
                         ga_lo,                  // global_addr[31:0]
                         ga_hi | 0x80000000u };  // global_addr[56:32] | type=2<<30

            i32x8 g1;
            g1[0] = 0x00020000;                               // wg_mask=0, data_size=4B, no pad/iter
            g1[1] = (int)((tdim0 & 0xFFFFu) << 16);           // barrier_addr=0 | tensor_dim0 lo
            g1[2] = (int)((tdim0 >> 16) | ((tdim1 & 0xFFFFu) << 16));
            g1[3] = (int)((tdim1 >> 16) | (ET << 16));        // tile_dim0 = 14
            g1[4] = (int)(ET | (CH << 16));                   // tile_dim1 = 14, tile_dim2 = 256
            g1[5] = WW;                                       // tensor_dim0_stride = 64 (lo 32)
            g1[6] = (int)((unsigned)(HH * WW) << 16);         // stride0 hi=0 | stride1 lo16 (4096)
            g1[7] = 0;                                        // stride1 hi (4096>>16 == 0)

            i32x4 g2 = { CH, 0, 0, 0 };   // tensor_dim2 = 256; dim3/stride2/tile_dim3 = 0
            i32x4 g3 = { 0, 0, 0, 0 };
            i32x8 g4 = { 0, 0, 0, 0, 0, 0, 0, 0 };   // extra operand on this toolchain (zero-filled)

            __builtin_amdgcn_tensor_load_to_lds(g0, g1, g2, g3, g4, 0);
            __builtin_amdgcn_s_wait_tensorcnt(0);
        }
    } else {
        // ---- Border tiles: scalar staging with explicit zero padding,
        //      same channel-major layout. One channel per thread.
        const int c = t;
        const float* xc = xb + (size_t)c * HH * WW;
        float* kc = keys + c * MKEYS;
        int m = 0;
        for (int my = 0; my < ET; ++my) {
            const int gy = gy0 + my;
            const float* xr = xc + gy * WW;
            #pragma unroll
            for (int mxp = 0; mxp < ET; ++mxp, ++m) {
                const int gxp = gx0 + mxp;
                float v = 0.f;
                if ((unsigned)gy < (unsigned)HH && (unsigned)gxp < (unsigned)WW)
                    v = xr[gxp];
                kc[m] = v;
            }
        }
    }
    __syncthreads();

    // ---- GEMM1: S[m,n] = sum_c keys[c][m] * keys[c][center(n)]
    //      13 M-tiles x 4 N-tiles = 52 WMMA tiles, K-loop = 256/4 steps.
    //      (M-tile 12 rows 196..207 produce garbage S -> masked to 0 later.)
    {
        const int mrow = lane & 15;
        const int koff = (lane < 16) ? 0 : 2;     // f32 16x16x4 A/B K-striping
        for (int tile = wid; tile < 52; tile += NWAVES) {
            const int mt = tile % 13;
            const int nt = tile / 13;
            const int n  = nt * 16 + mrow;        // query 0..63
            const int qy = n >> 3, qx = n & 7;
            const int mq = (qy + PADW) * ET + (qx + PADW);   // center key of query n
            const int am = mt * 16 + mrow;                    // A-fragment key row
            const float* colA = keys + am + koff * MKEYS;
            const float* colB = keys + mq + koff * MKEYS;
            v8f acc = {};
            #pragma unroll 8
            for (int c0 = 0; c0 < CH; c0 += 4) {
                v2f a, bb;
                a.x  = colA[(c0    ) * MKEYS];
                a.y  = colA[(c0 + 1) * MKEYS];
                bb.x = colB[(c0    ) * MKEYS];
                bb.y = colB[(c0 + 1) * MKEYS];
                acc = __builtin_amdgcn_wmma_f32_16x16x4_f32(
                          false, a, false, bb, (short)0, acc, false, false);
            }
            const int row0 = mt * 16 + ((lane < 16) ? 0 : 8);
            const int col  = nt * 16 + mrow;
            #pragma unroll
            for (int i = 0; i < 8; ++i)
                sp[(row0 + i) * SSTRIDE + col] = acc[i];
        }
    }
    __syncthreads();

    // ---- Softmax phase A: per-query max & 1/sum over 49 in-window taps
    if (t < NQ) {
        const int n  = t;
        const int qy = n >> 3, qx = n & 7;
        float mval = -3.402823466e38f;
        #pragma unroll
        for (int dy = 0; dy < KS; ++dy)
            #pragma unroll
            for (int dx = 0; dx < KS; ++dx) {
                const int mm = (qy + dy) * ET + (qx + dx);
                mval = fmaxf(mval, sp[mm * SSTRIDE + n]);
            }
        float sum = 0.f;
        #pragma unroll
        for (int dy = 0; dy < KS; ++dy)
            #pragma unroll
            for (int dx = 0; dx < KS; ++dx) {
                const int mm = (qy + dy) * ET + (qx + dx);
                sum += __expf(sp[mm * SSTRIDE + n] - mval);
            }
        mxv[n]  = mval;
        rsum[n] = 1.f / sum;
    }
    __syncthreads();

    // ---- Softmax phase B: S -> P in place; zero all out-of-window entries
    //      (incl. pad rows 196..207) so GEMM2 can accumulate all 208 rows.
    {
        #pragma unroll 4
        for (int k = 0; k < (MPAD * NQ) / NTHREADS; ++k) {       // 52 iters
            const int idx = k * NTHREADS + t;
            const int m   = idx >> 6;
            const int n   = idx & 63;
            const int qy  = n >> 3, qx = n & 7;
            float p = 0.f;
            if (m < MKEYS) {
                const int my  = m / ET;
                const int mxp = m - my * ET;
                const int dy  = my - qy, dx = mxp - qx;
                if ((unsigned)dy < (unsigned)KS && (unsigned)dx < (unsigned)KS)
                    p = __expf(sp[m * SSTRIDE + n] - mxv[n]) * rsum[n];
            }
            sp[m * SSTRIDE + n] = p;
        }
    }
    __syncthreads();

    // ---- GEMM2: Out[c,n] = sum_m keys[c][m] * P[m,n]
    //      16 C-tiles x 4 N-tiles = 64 WMMA tiles (8 per wave), K-loop = 208/4.
    //      A-fragment m-pairs are contiguous in channel-major LDS -> b64 loads.
    {
        const int mrow = lane & 15;
        const int koff = (lane < 16) ? 0 : 2;
        for (int j = 0; j < 8; ++j) {
            const int tile = wid * 8 + j;
            const int ct = tile >> 2;             // channel tile 0..15
            const int nt = tile & 3;              // query tile 0..3
            const int c  = ct * 16 + mrow;
            const int n  = nt * 16 + mrow;
            const float* arow = keys + c * MKEYS + koff;
            const float* brow = sp + n + koff * SSTRIDE;
            v8f acc = {};
            #pragma unroll 4
            for (int m0 = 0; m0 < MPAD; m0 += 4) {
                v2f a = *(const v2f*)(arow + m0);         // contiguous m-pair
                v2f bb;
                bb.x = brow[(m0    ) * SSTRIDE];
                bb.y = brow[(m0 + 1) * SSTRIDE];
                acc = __builtin_amdgcn_wmma_f32_16x16x4_f32(
                          false, a, false, bb, (short)0, acc, false, false);
            }
            const int row0 = ct * 16 + ((lane < 16) ? 0 : 8);    // channel
            const int col  = nt * 16 + mrow;                     // query
            const int qy = col >> 3, qx = col & 7;
            const int gy = ty * QT + qy;
            const int gxp = tx * QT + qx;
            float* op = out + (size_t)b * CH * HH * WW + gy * WW + gxp;
            #pragma unroll
            for (int i = 0; i < 8; ++i)
                op[(size_t)(row0 + i) * HH * WW] = acc[i];
        }
    }
}

extern "C" void kernel_launch(void* const* d_in, const int* in_sizes, int n_in,
                              void* d_out, int out_size, void* d_ws, size_t ws_size,
                              hipStream_t stream) {
    const float* x = (const float*)d_in[0];
    float* out = (float*)d_out;
    (void)in_sizes; (void)n_in; (void)out_size; (void)d_ws; (void)ws_size;

    const size_t shmem = (size_t)SMEM_FLOATS * sizeof(float);   // 257,792 B < 320 KB WGP LDS
    (void)hipFuncSetAttribute((const void*)local_attn_wmma_f32,
                              hipFuncAttributeMaxDynamicSharedMemorySize,
                              (int)shmem);

    dim3 grid(WW / QT, HH / QT, BATCH);    // 8 x 8 x 4 = 256 workgroups
    local_attn_wmma_f32<<<grid, NTHREADS, shmem, stream>>>(x, out);
}